// GeometryEncoder_47193100648596
// MI455X (gfx1250) — compile-verified
//
#include <hip/hip_runtime.h>
#include <math.h>

#define NPTS   8192
#define Bc     4
#define GYc    32
#define GXc    128
#define GZc    16
#define RAD    3
#define KTOT   (Bc*GYc*GXc*GZc)   // 262144
#define FILLKEY KTOT
#define EPSBN  1e-5f
#define TILE_J 2048

typedef __attribute__((ext_vector_type(2))) float v2f;
typedef __attribute__((ext_vector_type(8))) float v8f;

// ---------------------------------------------------------------------------
// 1) Geometry: all-pairs masked moments + closed-form 3x3 eigh -> geo_in[N,9]
// ---------------------------------------------------------------------------
__global__ __launch_bounds__(256) void geom_kernel(
    const float* __restrict__ voxels, const int* __restrict__ coors,
    float* __restrict__ geo_in)
{
    __shared__ float sx[TILE_J], sy[TILE_J], sz[TILE_J];
    __shared__ int   sb[TILE_J], syv[TILE_J], sxv[TILE_J];

    const int i = blockIdx.x * 256 + threadIdx.x;
    const float cx = voxels[i*3+0], cy = voxels[i*3+1], cz = voxels[i*3+2];
    const int bi = coors[i*4+0], gy = coors[i*4+1], gx = coors[i*4+2];

    float cnt=0.f, s1x=0.f, s1y=0.f, s1z=0.f;
    float mxx=0.f, mxy=0.f, mxz=0.f, myy=0.f, myz=0.f, mzz=0.f;

    for (int t = 0; t < NPTS; t += TILE_J) {
        __syncthreads();
        for (int j = threadIdx.x; j < TILE_J; j += 256) {
            sx[j]  = voxels[(t+j)*3+0];
            sy[j]  = voxels[(t+j)*3+1];
            sz[j]  = voxels[(t+j)*3+2];
            sb[j]  = coors[(t+j)*4+0];
            syv[j] = coors[(t+j)*4+1];
            sxv[j] = coors[(t+j)*4+2];
        }
        __syncthreads();
        #pragma unroll 4
        for (int j = 0; j < TILE_J; ++j) {
            int dy = syv[j] - gy; dy = dy < 0 ? -dy : dy;
            int dx = sxv[j] - gx; dx = dx < 0 ? -dx : dx;
            bool cond = (sb[j] == bi) & (dy <= RAD) & (dx <= RAD) & ((t + j) != i);
            float m = cond ? 1.0f : 0.0f;
            float px = sx[j], py = sy[j], pz = sz[j];
            cnt += m;
            s1x += m*px;    s1y += m*py;    s1z += m*pz;
            mxx += m*px*px; mxy += m*px*py; mxz += m*px*pz;
            myy += m*py*py; myz += m*py*pz; mzz += m*pz*pz;
        }
    }

    // covariance (same algebra as reference)
    const float ncl = fmaxf(cnt, 1.0f);
    float a00 = (mxx - 2.f*cx*s1x + cnt*cx*cx) / ncl;
    float a11 = (myy - 2.f*cy*s1y + cnt*cy*cy) / ncl;
    float a22 = (mzz - 2.f*cz*s1z + cnt*cz*cz) / ncl;
    float a01 = (mxy - cx*s1y - s1x*cy + cnt*cx*cy) / ncl;
    float a02 = (mxz - cx*s1z - s1x*cz + cnt*cx*cz) / ncl;
    float a12 = (myz - cy*s1z - s1y*cz + cnt*cy*cz) / ncl;

    const bool valid = (cnt >= 3.0f);
    if (!valid) { a00=1.f; a11=2.f; a22=3.f; a01=0.f; a02=0.f; a12=0.f; }

    // analytic symmetric 3x3 eigenvalues, ascending l0 <= l1 <= l2
    float l0, l1, l2;
    const float p1 = a01*a01 + a02*a02 + a12*a12;
    const float q  = (a00 + a11 + a22) * (1.f/3.f);
    if (p1 < 1e-30f) {
        l0=a00; l1=a11; l2=a22;
        float tswap;
        if (l0>l1){tswap=l0;l0=l1;l1=tswap;}
        if (l1>l2){tswap=l1;l1=l2;l2=tswap;}
        if (l0>l1){tswap=l0;l0=l1;l1=tswap;}
    } else {
        float d0=a00-q, d1=a11-q, d2=a22-q;
        float p2 = d0*d0 + d1*d1 + d2*d2 + 2.f*p1;
        float p  = sqrtf(p2 * (1.f/6.f));
        float ip = 1.f / p;
        float b00=d0*ip, b11=d1*ip, b22=d2*ip;
        float b01=a01*ip, b02=a02*ip, b12=a12*ip;
        float detB = b00*(b11*b22 - b12*b12)
                   - b01*(b01*b22 - b12*b02)
                   + b02*(b01*b12 - b11*b02);
        float r = 0.5f * detB;
        r = fminf(1.f, fmaxf(-1.f, r));
        float phi = acosf(r) * (1.f/3.f);
        l2 = q + 2.f*p*cosf(phi);
        l0 = q + 2.f*p*cosf(phi + 2.0943951023931953f); // + 2*pi/3
        l1 = 3.f*q - l0 - l2;
    }

    // eigenvector of smallest eigenvalue l0: cross products of rows of (A-l0*I)
    float r0x=a00-l0, r0y=a01,     r0z=a02;
    float r1x=a01,    r1y=a11-l0,  r1z=a12;
    float r2x=a02,    r2y=a12,     r2z=a22-l0;
    float c0x=r0y*r1z-r0z*r1y, c0y=r0z*r1x-r0x*r1z, c0z=r0x*r1y-r0y*r1x;
    float c1x=r0y*r2z-r0z*r2y, c1y=r0z*r2x-r0x*r2z, c1z=r0x*r2y-r0y*r2x;
    float c2x=r1y*r2z-r1z*r2y, c2y=r1z*r2x-r1x*r2z, c2z=r1x*r2y-r1y*r2x;
    float n0 = c0x*c0x+c0y*c0y+c0z*c0z;
    float n1 = c1x*c1x+c1y*c1y+c1z*c1z;
    float n2 = c2x*c2x+c2y*c2y+c2z*c2z;
    float nx=c0x, ny=c0y, nz=c0z, nb=n0;
    if (n1 > nb) { nx=c1x; ny=c1y; nz=c1z; nb=n1; }
    if (n2 > nb) { nx=c2x; ny=c2y; nz=c2z; nb=n2; }
    if (nb < 1e-30f) { nx=1.f; ny=0.f; nz=0.f; nb=1.f; }
    float inv = rsqrtf(nb);
    nx*=inv; ny*=inv; nz*=inv;
    if (nx*cx + ny*cy + nz*cz > 0.f) { nx=-nx; ny=-ny; nz=-nz; }
    if (!valid) { nx=0.f; ny=0.f; nz=0.f; }

    // curvature from |eigvals| sorted descending
    float e0=fabsf(l0), e1=fabsf(l1), e2=fabsf(l2), tsw;
    if (e0<e1){tsw=e0;e0=e1;e1=tsw;}
    if (e1<e2){tsw=e1;e1=e2;e2=tsw;}
    if (e0<e1){tsw=e0;e0=e1;e1=tsw;}
    float lam_sum = e0+e1+e2;
    bool ok = valid && (lam_sum > 1e-6f);
    float isum = 1.f / fmaxf(lam_sum, 1e-6f);
    float v0=e0*isum, v1=e1*isum, v2=e2*isum;
    float denom = v0 + 1e-6f;
    float k0 = (v0-v1)/denom, k1 = (v1-v2)/denom, k2 = v2/denom;
    if (!ok) { k0=0.f; k1=0.f; k2=0.f; }

    float* g = geo_in + (size_t)i*9;
    g[0]=cx; g[1]=cy; g[2]=cz;
    g[3]=nx; g[4]=ny; g[5]=nz;
    g[6]=k0; g[7]=k1; g[8]=k2;
}

// ---------------------------------------------------------------------------
// 2) WMMA f32 GEMM, compile-time specialized. One wave computes a 64x16
//    output tile (4 stacked 16x16 WMMA tiles sharing each B fragment).
//    K not a multiple of 4: tail loads are address-clamped (always in-bounds)
//    and value-selected to zero -> branchless, no EXEC manipulation.
// ---------------------------------------------------------------------------
template<int K, int Nc, bool BIAS>
__global__ __launch_bounds__(32) void gemm_wmma_f32(
    const float* __restrict__ X, const float* __restrict__ W,
    float* __restrict__ Y, const float* __restrict__ bias)
{
    constexpr int MT = 4;                 // 16-row tiles per wave
    const int ntN  = Nc >> 4;
    const int tm   = (blockIdx.x / ntN) * MT;
    const int tn   = blockIdx.x % ntN;
    const int lane = threadIdx.x;         // 0..31
    const int half = lane >> 4;           // 0 -> K {k,k+1}, 1 -> {k+2,k+3}
    const int l16  = lane & 15;
    const int colB = tn*16 + l16;

    v8f acc[MT];
    #pragma unroll
    for (int m = 0; m < MT; ++m) acc[m] = v8f{};

    #pragma unroll
    for (int k0 = 0; k0 < K; k0 += 4) {
        const int ka = k0 + 2*half;
        v2f b;
        if constexpr ((K & 3) == 0) {
            b.x = W[(size_t)(ka    )*Nc + colB];
            b.y = W[(size_t)(ka + 1)*Nc + colB];
        } else {
            // clamp address into range, select value to 0 for pad lanes
            const int kc0 = (ka     < K) ? ka     : (K - 1);
            const int kc1 = (ka + 1 < K) ? ka + 1 : (K - 1);
            float bx = W[(size_t)kc0*Nc + colB];
            float by = W[(size_t)kc1*Nc + colB];
            b.x = (ka     < K) ? bx : 0.0f;
            b.y = (ka + 1 < K) ? by : 0.0f;
        }
        #pragma unroll
        for (int m = 0; m < MT; ++m) {
            const int rowA = (tm + m)*16 + l16;
            v2f a;
            if constexpr ((K & 3) == 0) {
                // K multiple of 4 (and of 2): adjacent pair, 8B-aligned vector load
                a = *(const v2f*)(X + (size_t)rowA*K + ka);
            } else {
                const int kc0 = (ka     < K) ? ka     : (K - 1);
                const int kc1 = (ka + 1 < K) ? ka + 1 : (K - 1);
                float ax = X[(size_t)rowA*K + kc0];
                float ay = X[(size_t)rowA*K + kc1];
                a.x = (ka     < K) ? ax : 0.0f;
                a.y = (ka + 1 < K) ? ay : 0.0f;
            }
            acc[m] = __builtin_amdgcn_wmma_f32_16x16x4_f32(
                false, a, false, b, (short)0, acc[m], false, false);
        }
    }

    const float bv = BIAS ? bias[colB] : 0.0f;
    #pragma unroll
    for (int m = 0; m < MT; ++m) {
        #pragma unroll
        for (int r = 0; r < 8; ++r) {
            const int row = (tm + m)*16 + r + 8*half;
            Y[(size_t)row*Nc + colB] = acc[m][r] + bv;
        }
    }
}

// ---------------------------------------------------------------------------
// 3) BatchNorm helpers
// ---------------------------------------------------------------------------
__global__ __launch_bounds__(256) void colstats_kernel(
    const float* __restrict__ X, int Nc, float* __restrict__ mean, float* __restrict__ var)
{
    const int c = blockIdx.x;
    float s = 0.f, s2 = 0.f;
    for (int r = threadIdx.x; r < NPTS; r += 256) {
        float v = X[(size_t)r*Nc + c];
        s += v; s2 += v*v;
    }
    __shared__ float bs[256], bq[256];
    bs[threadIdx.x] = s; bq[threadIdx.x] = s2;
    __syncthreads();
    for (int off = 128; off > 0; off >>= 1) {
        if (threadIdx.x < off) {
            bs[threadIdx.x] += bs[threadIdx.x + off];
            bq[threadIdx.x] += bq[threadIdx.x + off];
        }
        __syncthreads();
    }
    if (threadIdx.x == 0) {
        float m = bs[0] * (1.0f / NPTS);
        mean[c] = m;
        var[c]  = bq[0] * (1.0f / NPTS) - m*m;
    }
}

// scale = g*rsqrt(var+eps); shift = beta - mean*scale  (tiny, one block)
__global__ __launch_bounds__(128) void bnparams_kernel(
    const float* __restrict__ g, const float* __restrict__ beta,
    const float* __restrict__ mean, const float* __restrict__ var,
    float* __restrict__ scale, float* __restrict__ shift, int n)
{
    int c = threadIdx.x;
    if (c < n) {
        float s = g[c] * rsqrtf(var[c] + EPSBN);
        scale[c] = s;
        shift[c] = beta[c] - mean[c]*s;
    }
}

// H = relu(X*scale + shift), float4-vectorized (Nc multiple of 4)
__global__ __launch_bounds__(256) void bnrelu_kernel(
    const float* __restrict__ X, float* __restrict__ H,
    const float* __restrict__ scale, const float* __restrict__ shift, int Nc)
{
    int i4 = blockIdx.x*blockDim.x + threadIdx.x;     // index in float4 units
    if (i4 >= (NPTS*Nc)/4) return;
    int c = (i4*4) & (Nc - 1);                        // Nc is a power of two
    float4 x  = ((const float4*)X)[i4];
    float4 sc = *(const float4*)(scale + c);
    float4 sh = *(const float4*)(shift + c);
    float4 h;
    h.x = fmaxf(fmaf(x.x, sc.x, sh.x), 0.f);
    h.y = fmaxf(fmaf(x.y, sc.y, sh.y), 0.f);
    h.z = fmaxf(fmaf(x.z, sc.z, sh.z), 0.f);
    h.w = fmaxf(fmaf(x.w, sc.w, sh.w), 0.f);
    ((float4*)H)[i4] = h;
}

// ---------------------------------------------------------------------------
// 4) unique/segment_max machinery
// ---------------------------------------------------------------------------
__global__ void zero_i32(int* p, int n) {
    int i = blockIdx.x*blockDim.x + threadIdx.x;
    if (i < n) p[i] = 0;
}

__global__ void mark_kernel(const int* __restrict__ coors, int* __restrict__ present) {
    int i = blockIdx.x*blockDim.x + threadIdx.x;
    if (i >= NPTS) return;
    int key = ((coors[i*4+0]*GYc + coors[i*4+1])*GXc + coors[i*4+2])*GZc + coors[i*4+3];
    present[key] = 1;
}

// block = 256 threads, 4 items/thread (1024/block), 256 blocks over KTOT
__global__ __launch_bounds__(256) void scan1_kernel(
    const int* __restrict__ present, int* __restrict__ rank, int* __restrict__ bsum)
{
    __shared__ int tmp[256];
    const int base = blockIdx.x * 1024;
    const int t = threadIdx.x;
    int v[4]; int s = 0;
    #pragma unroll
    for (int k = 0; k < 4; ++k) { v[k] = present[base + t*4 + k]; s += v[k]; }
    tmp[t] = s; __syncthreads();
    for (int off = 1; off < 256; off <<= 1) {
        int x = (t >= off) ? tmp[t - off] : 0;
        __syncthreads();
        tmp[t] += x;
        __syncthreads();
    }
    int run = tmp[t] - s;   // exclusive prefix for this thread's chunk
    #pragma unroll
    for (int k = 0; k < 4; ++k) { rank[base + t*4 + k] = run; run += v[k]; }
    if (t == 255) bsum[blockIdx.x] = tmp[255];
}

__global__ __launch_bounds__(256) void scan2_kernel(int* __restrict__ bsum, int* __restrict__ total)
{
    __shared__ int tmp[256];
    const int t = threadIdx.x;
    int v = bsum[t];
    tmp[t] = v; __syncthreads();
    for (int off = 1; off < 256; off <<= 1) {
        int x = (t >= off) ? tmp[t - off] : 0;
        __syncthreads();
        tmp[t] += x;
        __syncthreads();
    }
    bsum[t] = tmp[t] - v;   // exclusive
    if (t == 255) *total = tmp[255];
}

__global__ void scan3_kernel(int* __restrict__ rank, const int* __restrict__ bsum) {
    int i = blockIdx.x*blockDim.x + threadIdx.x;
    if (i < KTOT) rank[i] += bsum[i >> 10];
}

__global__ void uniqfill_kernel(const int* __restrict__ present, const int* __restrict__ rank,
                                int* __restrict__ uniq) {
    int k = blockIdx.x*blockDim.x + threadIdx.x;
    if (k < KTOT && present[k]) uniq[rank[k]] = k;
}

__global__ void init_vf_kernel(float* __restrict__ vf) {
    int i = blockIdx.x*blockDim.x + threadIdx.x;
    if (i < NPTS*128) vf[i] = -INFINITY;
}

__device__ inline void atomicMaxF(float* addr, float val) {
    if (val >= 0.0f) atomicMax((int*)addr, __float_as_int(val));
    else             atomicMin((unsigned int*)addr, __float_as_uint(val));
}

// one block per point, one thread per channel
__global__ __launch_bounds__(128) void scatter_max_kernel(
    const float* __restrict__ geo, const int* __restrict__ coors,
    const int* __restrict__ rank, float* __restrict__ vf)
{
    const int i = blockIdx.x;
    const int c = threadIdx.x;
    int key = ((coors[i*4+0]*GYc + coors[i*4+1])*GXc + coors[i*4+2])*GZc + coors[i*4+3];
    int r = rank[key];
    atomicMaxF(&vf[(size_t)r*128 + c], geo[(size_t)i*128 + c]);
}

// one block per output row: finitize voxel_feats + decode voxel_coors
__global__ __launch_bounds__(128) void finalize_kernel(
    float* __restrict__ vf, const int* __restrict__ uniq, const int* __restrict__ total,
    float* __restrict__ vcoors)
{
    const int r = blockIdx.x;
    const int U = *total;
    const int key = (r < U) ? uniq[r] : FILLKEY;
    if (threadIdx.x == 0) {
        int vb = key / (GYc*GXc*GZc);
        int vy = (key / (GXc*GZc)) % GYc;
        int vx = (key / GZc) % GXc;
        int vz = key % GZc;
        vcoors[r*4+0] = (float)vb;
        vcoors[r*4+1] = (float)vy;
        vcoors[r*4+2] = (float)vx;
        vcoors[r*4+3] = (float)vz;
    }
    float v = vf[(size_t)r*128 + threadIdx.x];
    if (!__builtin_isfinite(v)) v = 0.0f;
    vf[(size_t)r*128 + threadIdx.x] = v;
}

// ---------------------------------------------------------------------------
// launch
// ---------------------------------------------------------------------------
extern "C" void kernel_launch(void* const* d_in, const int* in_sizes, int n_in,
                              void* d_out, int out_size, void* d_ws, size_t ws_size,
                              hipStream_t stream)
{
    const float* voxels = (const float*)d_in[0];
    const int*   coors  = (const int*)  d_in[1];
    const float* W1 = (const float*)d_in[2];
    const float* g1 = (const float*)d_in[3];
    const float* b1 = (const float*)d_in[4];
    const float* W2 = (const float*)d_in[5];
    const float* g2 = (const float*)d_in[6];
    const float* b2 = (const float*)d_in[7];
    const float* W3 = (const float*)d_in[8];
    const float* b3 = (const float*)d_in[9];

    size_t off = 0;
    auto carve = [&](size_t bytes) -> char* {
        char* p = (char*)d_ws + off;
        off += (bytes + 255) & ~(size_t)255;
        return p;
    };
    float* geo_in = (float*)carve((size_t)NPTS*9*4);
    float* X1     = (float*)carve((size_t)NPTS*64*4);
    float* H1     = (float*)carve((size_t)NPTS*64*4);
    float* X2     = (float*)carve((size_t)NPTS*128*4);
    float* H2     = (float*)carve((size_t)NPTS*128*4);
    float* m1     = (float*)carve(64*4);
    float* v1     = (float*)carve(64*4);
    float* sc1    = (float*)carve(64*4);
    float* sh1    = (float*)carve(64*4);
    float* m2     = (float*)carve(128*4);
    float* v2     = (float*)carve(128*4);
    float* sc2    = (float*)carve(128*4);
    float* sh2    = (float*)carve(128*4);
    int*   present= (int*)  carve((size_t)KTOT*4);
    int*   rank   = (int*)  carve((size_t)KTOT*4);
    int*   bsum   = (int*)  carve(256*4);
    int*   uniq   = (int*)  carve((size_t)NPTS*4);
    int*   total  = (int*)  carve(4);

    float* geo_feats = (float*)d_out;                 // [8192,128]
    float* vfeats    = geo_feats + (size_t)NPTS*128;  // [8192,128]
    float* vcoors    = vfeats    + (size_t)NPTS*128;  // [8192,4] as float values

    // geometry -> geo_in
    geom_kernel<<<NPTS/256, 256, 0, stream>>>(voxels, coors, geo_in);

    // GEMM1: geo_in[8192,9] @ W1[9,64] -> X1
    gemm_wmma_f32<9, 64, false><<<(NPTS/64)*(64/16), 32, 0, stream>>>(geo_in, W1, X1, nullptr);

    // BN1 -> H1
    colstats_kernel<<<64, 256, 0, stream>>>(X1, 64, m1, v1);
    bnparams_kernel<<<1, 128, 0, stream>>>(g1, b1, m1, v1, sc1, sh1, 64);
    bnrelu_kernel<<<(NPTS*64/4 + 255)/256, 256, 0, stream>>>(X1, H1, sc1, sh1, 64);

    // GEMM2: H1[8192,64] @ W2[64,128] -> X2
    gemm_wmma_f32<64, 128, false><<<(NPTS/64)*(128/16), 32, 0, stream>>>(H1, W2, X2, nullptr);

    // BN2 -> H2
    colstats_kernel<<<128, 256, 0, stream>>>(X2, 128, m2, v2);
    bnparams_kernel<<<1, 128, 0, stream>>>(g2, b2, m2, v2, sc2, sh2, 128);
    bnrelu_kernel<<<(NPTS*128/4 + 255)/256, 256, 0, stream>>>(X2, H2, sc2, sh2, 128);

    // GEMM3: H2[8192,128] @ W3[128,128] + b3 -> geo_feats (output 0)
    gemm_wmma_f32<128, 128, true><<<(NPTS/64)*(128/16), 32, 0, stream>>>(H2, W3, geo_feats, b3);

    // unique + segment_max
    zero_i32<<<(KTOT+255)/256, 256, 0, stream>>>(present, KTOT);
    mark_kernel<<<NPTS/256, 256, 0, stream>>>(coors, present);
    scan1_kernel<<<KTOT/1024, 256, 0, stream>>>(present, rank, bsum);
    scan2_kernel<<<1, 256, 0, stream>>>(bsum, total);
    scan3_kernel<<<KTOT/256, 256, 0, stream>>>(rank, bsum);
    uniqfill_kernel<<<KTOT/256, 256, 0, stream>>>(present, rank, uniq);

    init_vf_kernel<<<(NPTS*128)/256, 256, 0, stream>>>(vfeats);
    scatter_max_kernel<<<NPTS, 128, 0, stream>>>(geo_feats, coors, rank, vfeats);
    finalize_kernel<<<NPTS, 128, 0, stream>>>(vfeats, uniq, total, vcoors);
}